// CausalSelfAttention_88854283420050
// MI455X (gfx1250) — compile-verified
//
#include <hip/hip_runtime.h>
#include <math.h>

// ---------------------------------------------------------------------------
// CausalSelfAttention forward for MI455X (gfx1250), fp32 via V_WMMA_F32_16X16X4_F32.
// T=2048, D=2048, H=16, C=128. ~100 GFLOP vs ~3us of HBM time at 23.3 TB/s ->
// compute-bound by ~100x, and the 67MB working set is GL2-resident (192MB L2).
// All GEMM-like stages run on the fp32 WMMA path; attention is flash-style so
// scores never touch HBM. Global->LDS staging is register double-buffered
// (one barrier per K-step, load latency hidden under WMMA) with
// global_prefetch_b8 two tiles ahead.
// ---------------------------------------------------------------------------

typedef float v2f __attribute__((ext_vector_type(2)));
typedef float v8f __attribute__((ext_vector_type(8)));

#define T_DIM 2048
#define D_DIM 2048
#define H_DIM 16
#define C_DIM 128
#define QKV_W 6144  // 3*D

static __device__ __forceinline__ v8f wmma_f32(v2f a, v2f b, v8f c) {
  // D = A(16x4) * B(4x16) + C(16x16), fp32.
  return __builtin_amdgcn_wmma_f32_16x16x4_f32(false, a, false, b, (short)0, c,
                                               false, false);
}

// ---------------------------------------------------------------------------
// Kernel 1/4: C[m][n] = sum_k A[m][k] * B[n][k] + bias[n]   (NT GEMM + bias)
// Tile 128(M) x 64(N) x 32(K), 256 threads = 8 waves in a 4x2 grid,
// each wave computes a 32x32 patch as 2x2 WMMA accumulators.
// Double-buffered LDS with register staging: one barrier per K-step.
// ---------------------------------------------------------------------------
__global__ __launch_bounds__(256) void gemm_nt_bias_kernel(
    const float* __restrict__ A, const float* __restrict__ B,
    const float* __restrict__ bias, float* __restrict__ C, int M, int N,
    int K) {
  __shared__ float As[2][128][36];  // stride 36: conflict-free A-frag reads
  __shared__ float Bs[2][32][72];   // stride 72: conflict-free B-frag reads

  const int tid = threadIdx.x;
  const int lane = tid & 31;
  const int wave = tid >> 5;
  const int row = lane & 15;  // M (A-frag) / N (B-frag, C/D) index
  const int hi = lane >> 4;   // selects K pair {2*hi, 2*hi+1}
  const int wm = wave >> 1;   // 0..3
  const int wn = wave & 1;    // 0..1
  const int m0 = blockIdx.y * 128;
  const int n0 = blockIdx.x * 64;

  v8f acc[2][2] = {};
  float4 ra[4];  // staged A tile slice (128x32 / 256 threads)
  float4 rb[2];  // staged B tile slice (64x32 / 256 threads)

  // Prologue: load tile 0 into registers, store to buffer 0.
#pragma unroll
  for (int it = 0; it < 4; ++it) {
    int f = tid + it * 256;
    int r = f >> 3, c4 = (f & 7) << 2;
    ra[it] = *(const float4*)(A + (size_t)(m0 + r) * K + c4);
  }
#pragma unroll
  for (int it = 0; it < 2; ++it) {
    int f = tid + it * 256;
    int r = f >> 3, c4 = (f & 7) << 2;
    rb[it] = *(const float4*)(B + (size_t)(n0 + r) * K + c4);
  }
#pragma unroll
  for (int it = 0; it < 4; ++it) {
    int f = tid + it * 256;
    int r = f >> 3, c4 = (f & 7) << 2;
    As[0][r][c4 + 0] = ra[it].x;
    As[0][r][c4 + 1] = ra[it].y;
    As[0][r][c4 + 2] = ra[it].z;
    As[0][r][c4 + 3] = ra[it].w;
  }
#pragma unroll
  for (int it = 0; it < 2; ++it) {
    int f = tid + it * 256;
    int r = f >> 3, c4 = (f & 7) << 2;
    Bs[0][c4 + 0][r] = rb[it].x;
    Bs[0][c4 + 1][r] = rb[it].y;
    Bs[0][c4 + 2][r] = rb[it].z;
    Bs[0][c4 + 3][r] = rb[it].w;
  }
  __syncthreads();

  const int nK = K >> 5;
  for (int i = 0; i < nK; ++i) {
    const int buf = i & 1;

    // Issue next tile's global loads first (latency hides under the WMMAs),
    // and prefetch the tile after that into GL2/WGP$.
    if (i + 1 < nK) {
      const int k0 = (i + 1) << 5;
#pragma unroll
      for (int it = 0; it < 4; ++it) {
        int f = tid + it * 256;
        int r = f >> 3, c4 = (f & 7) << 2;
        const float* p = A + (size_t)(m0 + r) * K + k0 + c4;
        if (i + 2 < nK) __builtin_prefetch(p + 32, 0, 3);
        ra[it] = *(const float4*)p;
      }
#pragma unroll
      for (int it = 0; it < 2; ++it) {
        int f = tid + it * 256;
        int r = f >> 3, c4 = (f & 7) << 2;
        const float* p = B + (size_t)(n0 + r) * K + k0 + c4;
        if (i + 2 < nK) __builtin_prefetch(p + 32, 0, 3);
        rb[it] = *(const float4*)p;
      }
    }

    // Compute from the current buffer.
#pragma unroll
    for (int kk = 0; kk < 32; kk += 4) {
      v2f a[2], b[2];
#pragma unroll
      for (int mi = 0; mi < 2; ++mi) {
        const float* p = &As[buf][wm * 32 + mi * 16 + row][kk + 2 * hi];
        v2f t;
        t.x = p[0];
        t.y = p[1];
        a[mi] = t;
      }
#pragma unroll
      for (int ni = 0; ni < 2; ++ni) {
        const int nb = wn * 32 + ni * 16 + row;
        v2f t;
        t.x = Bs[buf][kk + 2 * hi + 0][nb];
        t.y = Bs[buf][kk + 2 * hi + 1][nb];
        b[ni] = t;
      }
#pragma unroll
      for (int mi = 0; mi < 2; ++mi)
#pragma unroll
        for (int ni = 0; ni < 2; ++ni)
          acc[mi][ni] = wmma_f32(a[mi], b[ni], acc[mi][ni]);
    }

    // Dump staged registers into the other buffer; safe: nobody reads it
    // until after the barrier below.
    if (i + 1 < nK) {
      const int nb = buf ^ 1;
#pragma unroll
      for (int it = 0; it < 4; ++it) {
        int f = tid + it * 256;
        int r = f >> 3, c4 = (f & 7) << 2;
        As[nb][r][c4 + 0] = ra[it].x;
        As[nb][r][c4 + 1] = ra[it].y;
        As[nb][r][c4 + 2] = ra[it].z;
        As[nb][r][c4 + 3] = ra[it].w;
      }
#pragma unroll
      for (int it = 0; it < 2; ++it) {
        int f = tid + it * 256;
        int r = f >> 3, c4 = (f & 7) << 2;
        Bs[nb][c4 + 0][r] = rb[it].x;
        Bs[nb][c4 + 1][r] = rb[it].y;
        Bs[nb][c4 + 2][r] = rb[it].z;
        Bs[nb][c4 + 3][r] = rb[it].w;
      }
    }
    __syncthreads();
  }

  // Epilogue: C/D layout — N = lane&15, M = i + 8*(lane>>4) per VGPR i.
#pragma unroll
  for (int mi = 0; mi < 2; ++mi) {
#pragma unroll
    for (int ni = 0; ni < 2; ++ni) {
      int n = n0 + wn * 32 + ni * 16 + row;
      float bv = bias[n];
#pragma unroll
      for (int i = 0; i < 8; ++i) {
        int m = m0 + wm * 32 + mi * 16 + i + 8 * hi;
        C[(size_t)m * N + n] = acc[mi][ni][i] + bv;
      }
    }
  }
}

// ---------------------------------------------------------------------------
// Kernel 2: per-(head, t) LayerNorm over C=128 + RoPE, applied in place to the
// q and k sections of the qkv buffer. One wave per row; 4 elements per lane.
// ---------------------------------------------------------------------------
__global__ __launch_bounds__(256) void ln_rope_kernel(
    float* __restrict__ qkv, const float* __restrict__ qw,
    const float* __restrict__ kw) {
  const int lane = threadIdx.x & 31;
  const int wave = threadIdx.x >> 5;
  const int r = blockIdx.x * 8 + wave;  // 0 .. H*T-1
  const int h = r >> 11;                // / T
  const int t = r & (T_DIM - 1);

  // RoPE: element c uses table index c/2; pair0 covers c = 4*lane, 4*lane+1
  // (table idx 2*lane, exponent 4*lane/128), pair1 covers c+2, c+3.
  const float ln1e4 = 9.210340371976184f;  // log(10000)
  float inv0 = __expf(-((float)(4 * lane)) * (1.0f / 128.0f) * ln1e4);
  float inv1 = __expf(-((float)(4 * lane + 2)) * (1.0f / 128.0f) * ln1e4);
  float ang0 = (float)t * inv0;
  float ang1 = (float)t * inv1;
  float s0 = sinf(ang0), c0 = cosf(ang0);
  float s1 = sinf(ang1), c1 = cosf(ang1);

  for (int qk = 0; qk < 2; ++qk) {
    float* p = qkv + (size_t)t * QKV_W + qk * D_DIM + h * C_DIM + lane * 4;
    float4 v = *(const float4*)p;
    float s = v.x + v.y + v.z + v.w;
    float ss = v.x * v.x + v.y * v.y + v.z * v.z + v.w * v.w;
#pragma unroll
    for (int msk = 16; msk >= 1; msk >>= 1) {
      s += __shfl_xor(s, msk, 32);
      ss += __shfl_xor(ss, msk, 32);
    }
    float mean = s * (1.0f / 128.0f);
    float var = ss * (1.0f / 128.0f) - mean * mean;
    float rstd = rsqrtf(var + 1e-6f);

    const float* w = qk ? kw : qw;
    float4 wv = *(const float4*)(w + lane * 4);
    float e0 = (v.x - mean) * rstd * wv.x;
    float e1 = (v.y - mean) * rstd * wv.y;
    float e2 = (v.z - mean) * rstd * wv.z;
    float e3 = (v.w - mean) * rstd * wv.w;

    float4 o;
    o.x = e0 * c0 - e1 * s0;
    o.y = e1 * c0 + e0 * s0;
    o.z = e2 * c1 - e3 * s1;
    o.w = e3 * c1 + e2 * s1;
    *(float4*)p = o;
  }
}

// ---------------------------------------------------------------------------
// Kernel 3: flash attention, fp32 WMMA. Block = one head x 128 q rows,
// 8 waves; each wave independently owns 16 q rows (Q frags + O accumulators +
// per-row running max/sum in registers). K/V tiles (16x128) double-buffered
// in LDS with register staging + prefetch two tiles ahead.
// ---------------------------------------------------------------------------
__global__ __launch_bounds__(256) void attn_kernel(
    const float* __restrict__ qkv, float* __restrict__ attnO) {
  const int h = blockIdx.y;
  const int qBase = blockIdx.x * 128;
  const int tid = threadIdx.x;
  const int lane = tid & 31;
  const int wv = tid >> 5;
  const int row = lane & 15;
  const int hi = lane >> 4;

  __shared__ float Ks[2][16][132];  // stride 132: conflict-free QK^T B-frags
  __shared__ float Vs[2][16][136];  // stride 136: conflict-free PV B-frags
  __shared__ float Ps[8][16][20];   // per-wave P bounce (D-layout -> A-layout)

  const int qRow0 = qBase + wv * 16;
  const float scale = 0.08838834764831845f;  // 1/sqrt(C)

  // Preload this wave's 16x128 Q block as A-fragments (32 x v2f = 64 VGPRs),
  // pre-scaled by 1/sqrt(C).
  v2f qf[32];
  {
    const float* Qp = qkv + (size_t)(qRow0 + row) * QKV_W + h * C_DIM;
#pragma unroll
    for (int kk = 0; kk < 32; ++kk) {
      v2f t;
      t.x = Qp[kk * 4 + 2 * hi] * scale;
      t.y = Qp[kk * 4 + 2 * hi + 1] * scale;
      qf[kk] = t;
    }
  }

  float mrun[8], lrun[8];
  v8f o[8] = {};
#pragma unroll
  for (int i = 0; i < 8; ++i) {
    mrun[i] = -INFINITY;
    lrun[i] = 0.0f;
  }

  const int nTiles = (qBase >> 4) + 8;  // keys 0 .. qBase+127
  float4 kreg[2], vreg[2];

  // Prologue: stage tile 0.
#pragma unroll
  for (int it = 0; it < 2; ++it) {
    int f = tid + it * 256;
    int r = f >> 5, c4 = (f & 31) << 2;
    const float* kp = qkv + (size_t)r * QKV_W + D_DIM + h * C_DIM + c4;
    kreg[it] = *(const float4*)kp;
    vreg[it] = *(const float4*)(kp + D_DIM);
  }
#pragma unroll
  for (int it = 0; it < 2; ++it) {
    int f = tid + it * 256;
    int r = f >> 5, c4 = (f & 31) << 2;
    Ks[0][r][c4 + 0] = kreg[it].x;
    Ks[0][r][c4 + 1] = kreg[it].y;
    Ks[0][r][c4 + 2] = kreg[it].z;
    Ks[0][r][c4 + 3] = kreg[it].w;
    Vs[0][r][c4 + 0] = vreg[it].x;
    Vs[0][r][c4 + 1] = vreg[it].y;
    Vs[0][r][c4 + 2] = vreg[it].z;
    Vs[0][r][c4 + 3] = vreg[it].w;
  }
  __syncthreads();

  for (int j = 0; j < nTiles; ++j) {
    const int buf = j & 1;
    const int kv0 = j << 4;

    // Issue next tile's loads (latency hides under the WMMAs below).
    if (j + 1 < nTiles) {
      const int kv1 = (j + 1) << 4;
#pragma unroll
      for (int it = 0; it < 2; ++it) {
        int f = tid + it * 256;
        int r = f >> 5, c4 = (f & 31) << 2;
        const float* kp =
            qkv + (size_t)(kv1 + r) * QKV_W + D_DIM + h * C_DIM + c4;
        if (j + 2 < nTiles) {
          __builtin_prefetch(kp + 16 * QKV_W, 0, 3);
          __builtin_prefetch(kp + 16 * QKV_W + D_DIM, 0, 3);
        }
        kreg[it] = *(const float4*)kp;
        vreg[it] = *(const float4*)(kp + D_DIM);
      }
    }

    if (kv0 <= qRow0) {  // causal: this wave needs this key tile
      // S(16x16) = Qs . K^T ; B element (k=c, n=key) = Ks[key][c].
      v8f s = {};
#pragma unroll
      for (int kk = 0; kk < 32; ++kk) {
        v2f b;
        b.x = Ks[buf][row][kk * 4 + 2 * hi];
        b.y = Ks[buf][row][kk * 4 + 2 * hi + 1];
        s = wmma_f32(qf[kk], b, s);
      }
      // Causal mask on the diagonal tile (kv0 == qRow0).
      if (kv0 + 15 > qRow0) {
#pragma unroll
        for (int i = 0; i < 8; ++i) {
          int m = i + 8 * hi;                           // q row within tile
          if (kv0 + row > qRow0 + m) s[i] = -INFINITY;  // N = lane&15
        }
      }
      // Online softmax: row stats reduce across the 16 lanes of each half.
#pragma unroll
      for (int i = 0; i < 8; ++i) {
        float mx = s[i];
#pragma unroll
        for (int msk = 8; msk >= 1; msk >>= 1)
          mx = fmaxf(mx, __shfl_xor(mx, msk, 32));
        float mnew = fmaxf(mrun[i], mx);
        float alpha = __expf(mrun[i] - mnew);
        float pv = __expf(s[i] - mnew);
        float rs = pv;
#pragma unroll
        for (int msk = 8; msk >= 1; msk >>= 1) rs += __shfl_xor(rs, msk, 32);
        lrun[i] = lrun[i] * alpha + rs;
        mrun[i] = mnew;
#pragma unroll
        for (int c = 0; c < 8; ++c) o[c][i] *= alpha;
        Ps[wv][i + 8 * hi][row] = pv;  // wave-private; no barrier needed
      }
      // O += P(16x16) . V(16x128): A-frags from Ps, B-frags from Vs.
      v2f a[4];
#pragma unroll
      for (int kk = 0; kk < 4; ++kk) {
        v2f t;
        t.x = Ps[wv][row][kk * 4 + 2 * hi];
        t.y = Ps[wv][row][kk * 4 + 2 * hi + 1];
        a[kk] = t;
      }
#pragma unroll
      for (int c = 0; c < 8; ++c) {
#pragma unroll
        for (int kk = 0; kk < 4; ++kk) {
          v2f b;
          b.x = Vs[buf][kk * 4 + 2 * hi + 0][c * 16 + row];
          b.y = Vs[buf][kk * 4 + 2 * hi + 1][c * 16 + row];
          o[c] = wmma_f32(a[kk], b, o[c]);
        }
      }
    }

    // Dump staged K/V into the other buffer (read only after the barrier).
    if (j + 1 < nTiles) {
      const int nb = buf ^ 1;
#pragma unroll
      for (int it = 0; it < 2; ++it) {
        int f = tid + it * 256;
        int r = f >> 5, c4 = (f & 31) << 2;
        Ks[nb][r][c4 + 0] = kreg[it].x;
        Ks[nb][r][c4 + 1] = kreg[it].y;
        Ks[nb][r][c4 + 2] = kreg[it].z;
        Ks[nb][r][c4 + 3] = kreg[it].w;
        Vs[nb][r][c4 + 0] = vreg[it].x;
        Vs[nb][r][c4 + 1] = vreg[it].y;
        Vs[nb][r][c4 + 2] = vreg[it].z;
        Vs[nb][r][c4 + 3] = vreg[it].w;
      }
    }
    __syncthreads();
  }

  // Normalize and write: out[t][h*C + c] (ready for the projection GEMM).
#pragma unroll
  for (int c = 0; c < 8; ++c) {
#pragma unroll
    for (int i = 0; i < 8; ++i) {
      int t = qRow0 + i + 8 * hi;
      int d = h * C_DIM + c * 16 + row;
      attnO[(size_t)t * D_DIM + d] = o[c][i] / lrun[i];
    }
  }
}

// ---------------------------------------------------------------------------
extern "C" void kernel_launch(void* const* d_in, const int* in_sizes, int n_in,
                              void* d_out, int out_size, void* d_ws,
                              size_t ws_size, hipStream_t stream) {
  (void)in_sizes;
  (void)n_in;
  (void)out_size;
  (void)ws_size;
  const float* x = (const float*)d_in[0];
  const float* W_attn = (const float*)d_in[1];
  const float* b_attn = (const float*)d_in[2];
  const float* W_proj = (const float*)d_in[3];
  const float* b_proj = (const float*)d_in[4];
  const float* q_ln_w = (const float*)d_in[5];
  const float* k_ln_w = (const float*)d_in[6];
  float* out = (float*)d_out;

  float* qkv = (float*)d_ws;                   // T * 3D floats (50.3 MB)
  float* attnO = qkv + (size_t)T_DIM * QKV_W;  // T * D floats  (16.8 MB)

  dim3 blk(256);

  // 1) qkv = x @ W_attn^T + b_attn
  gemm_nt_bias_kernel<<<dim3(QKV_W / 64, T_DIM / 128), blk, 0, stream>>>(
      x, W_attn, b_attn, qkv, T_DIM, QKV_W, D_DIM);

  // 2) per-head LayerNorm + RoPE on q and k, in place.
  ln_rope_kernel<<<dim3((H_DIM * T_DIM) / 8), blk, 0, stream>>>(qkv, q_ln_w,
                                                                k_ln_w);

  // 3) causal flash attention -> attnO[t][h*C+c]
  attn_kernel<<<dim3(T_DIM / 128, H_DIM), blk, 0, stream>>>(qkv, attnO);

  // 4) out = attnO @ W_proj^T + b_proj
  gemm_nt_bias_kernel<<<dim3(D_DIM / 64, T_DIM / 128), blk, 0, stream>>>(
      attnO, W_proj, b_proj, out, T_DIM, D_DIM, D_DIM);
}